// VQ_EMA_17566416241064
// MI455X (gfx1250) — compile-verified
//
#include <hip/hip_runtime.h>
#include <hip/hip_bf16.h>

typedef __attribute__((ext_vector_type(2))) float v2f;
typedef __attribute__((ext_vector_type(8))) float v8f;

#define B_SZ 1024
#define D_SZ 64
#define L_SZ 64
#define K_SZ 512
#define DECAY 0.99f
#define EPSV 1e-6f

// ---------------------------------------------------------------------------
// Kernel A: e2[k,l] = sum_d emb[d,k,l]^2   (coalesced over k*L+l)
// ---------------------------------------------------------------------------
__global__ void e2_kernel(const float* __restrict__ emb, float* __restrict__ e2) {
  int i = blockIdx.x * blockDim.x + threadIdx.x;   // over K*L = 32768
  float s = 0.0f;
  #pragma unroll 4
  for (int d = 0; d < D_SZ; ++d) {
    float v = emb[(size_t)d * (K_SZ * L_SZ) + i];
    s += v * v;
  }
  e2[i] = s;
}

// ---------------------------------------------------------------------------
// Kernel B: xe GEMM via V_WMMA_F32_16X16X4_F32 + fused argmin over K.
// One wave handles 16 rows of B for one l. 4 waves/block -> 64 rows/block.
// A (16x4 f32) layout: lane(0..15)=M, v.x=K0/K1 (lo half) or K2/K3 (hi half).
// C (16x16 f32) layout: VGPR j -> M=j (lanes0-15) / M=j+8 (lanes16-31), N=lane&15.
// ---------------------------------------------------------------------------
__global__ __launch_bounds__(128) void gemm_argmin_kernel(
    const float* __restrict__ x,      // (B, D, L)
    const float* __restrict__ emb,    // (D, K, L)
    const float* __restrict__ e2,     // (K, L)
    int* __restrict__ idx_out)        // (B, L)
{
  const int l     = blockIdx.x & (L_SZ - 1);
  const int btile = blockIdx.x >> 6;
  const int wave  = threadIdx.x >> 5;
  const int lane  = threadIdx.x & 31;
  const int m16   = lane & 15;
  const bool hi   = lane >= 16;
  const int b0    = btile * 64 + wave * 16;

  // Preload the full A strip: 16 rows x 64 D, as 16 K=4 fragments.
  v2f a[16];
  const float* xrow = x + (size_t)(b0 + m16) * (D_SZ * L_SZ) + l;
  #pragma unroll
  for (int t = 0; t < 16; ++t) {
    int d0 = t * 4 + (hi ? 2 : 0);
    a[t].x = xrow[(d0 + 0) * L_SZ];
    a[t].y = xrow[(d0 + 1) * L_SZ];
  }

  float minv[8];
  int   mink[8];
  #pragma unroll
  for (int j = 0; j < 8; ++j) { minv[j] = 3.4e38f; mink[j] = 0; }

  for (int nt = 0; nt < 32; ++nt) {
    const int k = nt * 16 + m16;                        // this lane's column
    const float* bcol = emb + (size_t)k * L_SZ + l;     // emb[d,k,l]
    v8f c = {};
    #pragma unroll
    for (int t = 0; t < 16; ++t) {
      int d0 = t * 4 + (hi ? 2 : 0);
      v2f bf;
      bf.x = bcol[(size_t)(d0 + 0) * (K_SZ * L_SZ)];
      bf.y = bcol[(size_t)(d0 + 1) * (K_SZ * L_SZ)];
      c = __builtin_amdgcn_wmma_f32_16x16x4_f32(
              /*neg_a=*/false, a[t], /*neg_b=*/false, bf,
              /*c_mod=*/(short)0, c, /*reuse_a=*/false, /*reuse_b=*/false);
    }
    const float e2v = e2[k * L_SZ + l];
    #pragma unroll
    for (int j = 0; j < 8; ++j) {
      float d2 = e2v - 2.0f * c[j];
      if (d2 < minv[j] || (d2 == minv[j] && k < mink[j])) { minv[j] = d2; mink[j] = k; }
    }
  }

  // Min-reduce across the 16-lane half groups (wave32).
  #pragma unroll
  for (int off = 8; off >= 1; off >>= 1) {
    #pragma unroll
    for (int j = 0; j < 8; ++j) {
      float ov = __shfl_xor(minv[j], off, 32);
      int   ok = __shfl_xor(mink[j], off, 32);
      if (ov < minv[j] || (ov == minv[j] && ok < mink[j])) { minv[j] = ov; mink[j] = ok; }
    }
  }
  if (m16 == 0) {
    int mbase = b0 + (hi ? 8 : 0);
    #pragma unroll
    for (int j = 0; j < 8; ++j)
      idx_out[(size_t)(mbase + j) * L_SZ + l] = mink[j];
  }
}

// ---------------------------------------------------------------------------
// Kernel C: gather q, quantized, one-hot scatter, counts, dw_batch scatter,
// commitment-loss accumulation. Thread id = b*D*L + d*L + l (l contiguous).
// ---------------------------------------------------------------------------
__global__ __launch_bounds__(256) void scatter_kernel(
    const float* __restrict__ x, const float* __restrict__ emb,
    const int* __restrict__ idx,
    float* __restrict__ quantized, float* __restrict__ encoding,
    float* __restrict__ dwb, float* __restrict__ counts,
    float* __restrict__ loss_acc)
{
  int t = blockIdx.x * blockDim.x + threadIdx.x;  // over B*D*L = 4M
  int l = t & (L_SZ - 1);
  int d = (t >> 6) & (D_SZ - 1);
  int b = t >> 12;

  int k = idx[b * L_SZ + l];
  size_t xoff = (size_t)b * (D_SZ * L_SZ) + d * L_SZ + l;
  float xv = x[xoff];
  float qv = emb[(size_t)d * (K_SZ * L_SZ) + (size_t)k * L_SZ + l];
  float dq = qv - xv;
  quantized[xoff] = xv + dq;                       // straight-through
  atomicAdd(&dwb[(size_t)d * (K_SZ * L_SZ) + (size_t)k * L_SZ + l], xv);
  if (d == 0) {
    atomicAdd(&counts[k * L_SZ + l], 1.0f);
    encoding[(size_t)b * (K_SZ * L_SZ) + (size_t)k * L_SZ + l] = 1.0f;
  }

  // block-level reduction of (q-x)^2
  float v = dq * dq;
  #pragma unroll
  for (int off = 16; off >= 1; off >>= 1) v += __shfl_xor(v, off, 32);
  __shared__ float sbuf[8];
  int wv = threadIdx.x >> 5;
  if ((threadIdx.x & 31) == 0) sbuf[wv] = v;
  __syncthreads();
  if (threadIdx.x == 0) {
    float s = 0.0f;
    #pragma unroll
    for (int i = 0; i < 8; ++i) s += sbuf[i];
    atomicAdd(loss_acc, s);
  }
}

// ---------------------------------------------------------------------------
// Kernel D1: cs update. One block (64 threads) per k.
// ---------------------------------------------------------------------------
__global__ __launch_bounds__(64) void cs_kernel(
    const float* __restrict__ cluster_size, const float* __restrict__ counts,
    float* __restrict__ cs_out)
{
  int k = blockIdx.x;
  int l = threadIdx.x;
  float v = cluster_size[k * L_SZ + l];
  __shared__ float sb[64];
  sb[l] = v;
  __syncthreads();
  #pragma unroll
  for (int off = 32; off >= 1; off >>= 1) {
    if (l < off) sb[l] += sb[l + off];
    __syncthreads();
  }
  float n = sb[0] + 1e-6f;
  float cs = (v + EPSV) / (n + (float)K_SZ * EPSV) * n;
  cs = DECAY * cs + (1.0f - DECAY) * counts[k * L_SZ + l];
  cs_out[k * L_SZ + l] = cs;
}

// ---------------------------------------------------------------------------
// Kernel D2: dw_buf / emb_new elementwise over (D,K,L).
// ---------------------------------------------------------------------------
__global__ __launch_bounds__(256) void update_kernel(
    const float* __restrict__ dw, const float* __restrict__ emb,
    const float* __restrict__ dwb, const float* __restrict__ cs,
    float* __restrict__ dw_buf_out, float* __restrict__ emb_out)
{
  int i = blockIdx.x * blockDim.x + threadIdx.x;  // D*K*L = 2M
  int kl = i & (K_SZ * L_SZ - 1);
  float c = cs[kl];
  float db = DECAY * dw[i] + (1.0f - DECAY) * (dwb[i] / c);
  dw_buf_out[i] = db;
  emb_out[i] = DECAY * emb[i] + (1.0f - DECAY) * db;
}

// ---------------------------------------------------------------------------
// Kernel E: perplexity per l + finalize loss.
// ---------------------------------------------------------------------------
__global__ __launch_bounds__(64) void finalize_kernel(
    const float* __restrict__ counts, const float* __restrict__ loss_acc,
    float* __restrict__ perp_out, float* __restrict__ loss_out)
{
  int l = threadIdx.x;
  float s = 0.0f;
  for (int k = 0; k < K_SZ; ++k) {
    float p = counts[k * L_SZ + l] * (1.0f / (float)B_SZ);
    s += p * logf(p + 1e-10f);
  }
  perp_out[l] = expf(-s) * (1.0f / (float)K_SZ);
  if (l == 0)
    loss_out[0] = loss_acc[0] * (1.0f / (float)(B_SZ * D_SZ * L_SZ));
}

// ---------------------------------------------------------------------------
extern "C" void kernel_launch(void* const* d_in, const int* in_sizes, int n_in,
                              void* d_out, int out_size, void* d_ws, size_t ws_size,
                              hipStream_t stream) {
  const float* x            = (const float*)d_in[0];  // (B,D,h,w)
  const float* emb          = (const float*)d_in[1];  // (D,K,L)
  const float* dw           = (const float*)d_in[2];  // (D,K,L)
  const float* cluster_size = (const float*)d_in[3];  // (K,L)

  float* out = (float*)d_out;
  // output layout: quantized | encoding | loss | perplexity | emb_new | dw_buf | cs
  float* quantized = out;                               // 4,194,304
  float* encoding  = out + 4194304;                     // 33,554,432
  float* loss_out  = out + 4194304 + 33554432;          // 1
  float* perp_out  = loss_out + 1;                      // 64
  float* emb_out   = perp_out + 64;                     // 2,097,152
  float* dwbuf_out = emb_out + 2097152;                 // 2,097,152
  float* cs_out    = dwbuf_out + 2097152;               // 32,768

  float* ws       = (float*)d_ws;
  float* e2       = ws;                                 // 32,768
  float* counts   = ws + 32768;                         // 32,768
  float* dwb      = ws + 65536;                         // 2,097,152
  float* loss_acc = ws + 65536 + 2097152;               // 1
  int*   indices  = (int*)(ws + 65536 + 2097152 + 1);   // 65,536 ints

  // Zero the accumulation targets (graph-capture-safe async memsets).
  hipMemsetAsync(counts,   0, (size_t)32768 * sizeof(float), stream);
  hipMemsetAsync(dwb,      0, (size_t)2097152 * sizeof(float), stream);
  hipMemsetAsync(loss_acc, 0, sizeof(float), stream);
  hipMemsetAsync(encoding, 0, (size_t)33554432 * sizeof(float), stream);

  e2_kernel<<<(K_SZ * L_SZ) / 256, 256, 0, stream>>>(emb, e2);
  gemm_argmin_kernel<<<(B_SZ / 64) * L_SZ, 128, 0, stream>>>(x, emb, e2, indices);
  scatter_kernel<<<(B_SZ * D_SZ * L_SZ) / 256, 256, 0, stream>>>(
      x, emb, indices, quantized, encoding, dwb, counts, loss_acc);
  cs_kernel<<<K_SZ, 64, 0, stream>>>(cluster_size, counts, cs_out);
  update_kernel<<<(D_SZ * K_SZ * L_SZ) / 256, 256, 0, stream>>>(
      dw, emb, dwb, cs_out, dwbuf_out, emb_out);
  finalize_kernel<<<1, 64, 0, stream>>>(counts, loss_acc, perp_out, loss_out);
}